// ReRaLSTM_85435489452157
// MI455X (gfx1250) — compile-verified
//
#include <hip/hip_runtime.h>
#include <hip/hip_bf16.h>
#include <math.h>

// Problem constants (from reference): N=2048, T=16, D=16, H=64, R=64, 4H=256
#define NN 2048
#define TT 16
#define DD 16
#define HH 64
#define GG 256  // 4*H

typedef __attribute__((ext_vector_type(2))) float f32x2;
typedef __attribute__((ext_vector_type(8))) float f32x8;

// D = A(16x4) * B(4x16) + C(16x16), all f32  -> v_wmma_f32_16x16x4_f32
__device__ __forceinline__ f32x8 wmma4(f32x2 a, f32x2 b, f32x8 c) {
  return __builtin_amdgcn_wmma_f32_16x16x4_f32(
      /*neg_a=*/false, a, /*neg_b=*/false, b,
      /*c_mod=*/(short)0, c, /*reuse_a=*/false, /*reuse_b=*/false);
}

__device__ __forceinline__ float leaky(float x) { return fmaxf(0.2f * x, x); }
__device__ __forceinline__ float sigmoidf(float x) { return 1.f / (1.f + __expf(-x)); }

// ---------------------------------------------------------------------------
// Kernel 1: LSTM over T=16 steps. Block = 64 threads (2 waves), each wave
// owns a 16-row tile of N. Per step: gates[16,256] = bias + x_t@W_ih^T +
// h@W_hh^T via V_WMMA_F32_16X16X4_F32.  W_ih/W_hh B-fragments come straight
// from global (contiguous because weights are stored [4H, H] row-major).
// h lives in per-wave LDS [16][64]; gates round-trip through LDS so the
// elementwise i/f/g/o math needs no cross-lane data movement.
// ---------------------------------------------------------------------------
__global__ __launch_bounds__(64) void lstm_kernel(
    const float* __restrict__ x,     // [N][T][D]
    const float* __restrict__ W_ih,  // [4H][D]
    const float* __restrict__ W_hh,  // [4H][H]
    const float* __restrict__ b_ih,  // [4H]
    const float* __restrict__ b_hh,  // [4H]
    float* __restrict__ seq_emb)     // [N][H]
{
  __shared__ __align__(16) float hbuf[2][16 * HH];   // per-wave h state
  __shared__ __align__(16) float gates[2][16 * GG];  // per-wave gate scratch

  const int tid  = threadIdx.x;
  const int w    = tid >> 5;
  const int lane = tid & 31;
  const int m    = lane & 15;   // row within 16-row tile
  const int hi   = lane >> 4;   // half-wave selector (K offset 0 or 2)
  const int rb   = blockIdx.x * 32 + w * 16;

  float* hb = hbuf[w];
  float* gb = gates[w];

  for (int u = lane; u < 16 * HH; u += 32) hb[u] = 0.f;

  float cst[32];
#pragma unroll
  for (int u = 0; u < 32; ++u) cst[u] = 0.f;

  for (int t = 0; t < TT; ++t) {
    // A-fragments for x_t tile [16 x 16]: lane m holds rows, hi picks K pair
    f32x2 ax[4];
#pragma unroll
    for (int ks = 0; ks < 4; ++ks)
      ax[ks] = *(const f32x2*)(x + (size_t)(rb + m) * (TT * DD) + t * DD + ks * 4 + hi * 2);

    for (int ct = 0; ct < 16; ++ct) {  // 16 column tiles of the 256 gates
      const int col = ct * 16 + m;
      const float bv = b_ih[col] + b_hh[col];
      f32x8 acc = {bv, bv, bv, bv, bv, bv, bv, bv};

      // x @ W_ih^T : K = 16  (B frag = contiguous pair of W_ih[col][k..k+1])
#pragma unroll
      for (int ks = 0; ks < 4; ++ks) {
        f32x2 b = *(const f32x2*)(W_ih + col * DD + ks * 4 + hi * 2);
        acc = wmma4(ax[ks], b, acc);
      }
      // h @ W_hh^T : K = 64  (A frag from LDS h, B frag from global W_hh)
#pragma unroll
      for (int ks = 0; ks < 16; ++ks) {
        f32x2 a = *(const f32x2*)(hb + m * HH + ks * 4 + hi * 2);
        f32x2 b = *(const f32x2*)(W_hh + col * HH + ks * 4 + hi * 2);
        acc = wmma4(a, b, acc);
      }
      // C layout: VGPR r -> row (r + 8*hi), col = ct*16 + (lane&15)
#pragma unroll
      for (int r = 0; r < 8; ++r) gb[(r + 8 * hi) * GG + col] = acc[r];
    }

    // Elementwise gate math: 16*64 = 1024 elems -> 32 per lane.
#pragma unroll
    for (int u = 0; u < 32; ++u) {
      const int e = u * 32 + lane;
      const int row = e >> 6, cc = e & 63;
      float iv = sigmoidf(gb[row * GG + cc]);
      float fv = sigmoidf(gb[row * GG + 64 + cc]);
      float gv = tanhf(gb[row * GG + 128 + cc]);
      float ov = sigmoidf(gb[row * GG + 192 + cc]);
      float c = fv * cst[u] + iv * gv;
      cst[u] = c;
      hb[row * HH + cc] = ov * tanhf(c);  // A-layout for next step
    }
  }

  for (int u = lane; u < 16 * HH; u += 32) {
    const int row = u >> 6, cc = u & 63;
    seq_emb[(size_t)(rb + row) * HH + cc] = hb[u];
  }
}

// ---------------------------------------------------------------------------
// Kernel 2: ht[i] = leaky(seq@w2+b2) + leaky(seq@w3+b3)   (both index row i)
// ---------------------------------------------------------------------------
__global__ void ht_kernel(const float* __restrict__ seq,
                          const float* __restrict__ w2, const float* __restrict__ b2,
                          const float* __restrict__ w3, const float* __restrict__ b3,
                          float* __restrict__ ht)
{
  const int i = blockIdx.x * 256 + threadIdx.x;
  if (i >= NN) return;
  float a2 = 0.f, a3 = 0.f;
#pragma unroll 8
  for (int k = 0; k < HH; ++k) {
    const float s = seq[(size_t)i * HH + k];
    a2 += s * w2[k];
    a3 += s * w3[k];
  }
  ht[i] = leaky(a2 + b2[0]) + leaky(a3 + b3[0]);
}

// ---------------------------------------------------------------------------
// Kernel 3: the bandwidth king — stream relation [N,N,64] (1.07 GB) once.
// logits[i,j] = rel_mask[i,j] + ht[i] + leaky(dot(relation[i,j,:], w1) + b1)
// Each thread consumes one contiguous 256B row via 16 float4 loads.
// ---------------------------------------------------------------------------
__global__ void rel_kernel(const float* __restrict__ rel,
                           const float* __restrict__ mask,
                           const float* __restrict__ w1, const float* __restrict__ b1,
                           const float* __restrict__ ht,
                           float* __restrict__ logits)
{
  const size_t pair = (size_t)blockIdx.x * 256 + threadIdx.x;
  const int i = (int)(pair >> 11);
  const float4* r4 = (const float4*)(rel + pair * 64);
  const float4* w4p = (const float4*)w1;
  float acc = 0.f;
#pragma unroll
  for (int q = 0; q < 16; ++q) {
    const float4 rv = r4[q];
    const float4 wv = w4p[q];
    acc += rv.x * wv.x + rv.y * wv.y + rv.z * wv.z + rv.w * wv.w;
  }
  logits[pair] = mask[pair] + ht[i] + leaky(acc + b1[0]);
}

// ---------------------------------------------------------------------------
// Kernel 4: softmax over axis=0 (down the rows, per column). Thread-per-column,
// loads coalesced across consecutive columns. In-place on logits.
// ---------------------------------------------------------------------------
__global__ void softmax_kernel(float* __restrict__ logits)
{
  const int col = blockIdx.x * 256 + threadIdx.x;
  float mx = -3.4e38f;
  for (int i = 0; i < NN; ++i) mx = fmaxf(mx, logits[(size_t)i * NN + col]);
  float s = 0.f;
  for (int i = 0; i < NN; ++i) s += __expf(logits[(size_t)i * NN + col] - mx);
  const float inv = 1.f / s;
  for (int i = 0; i < NN; ++i) {
    const size_t idx = (size_t)i * NN + col;
    logits[idx] = __expf(logits[idx] - mx) * inv;
  }
}

// ---------------------------------------------------------------------------
// Kernel 5: outputs_proped = weight_masked[N,N] @ seq_emb[N,64] via WMMA f32.
// Block = 2 waves; wave owns 16 output rows x all 64 cols (4 C tiles).
// seq_emb is staged in LDS in (k-pair, col) layout in chunks of 128 K-rows so
// B fragments are single ds_load_b64s.
// ---------------------------------------------------------------------------
__global__ __launch_bounds__(64) void prop_kernel(
    const float* __restrict__ wm,   // [N][N] softmaxed weights
    const float* __restrict__ seq,  // [N][H]
    float* __restrict__ out)        // [N][H]
{
  __shared__ __align__(16) float pb[128 * HH];  // pb[kp*128 + col*2 + parity]

  const int tid  = threadIdx.x;
  const int w    = tid >> 5;
  const int lane = tid & 31;
  const int m    = lane & 15;
  const int hi   = lane >> 4;
  const int rb   = blockIdx.x * 32 + w * 16;

  const f32x8 z = {0.f, 0.f, 0.f, 0.f, 0.f, 0.f, 0.f, 0.f};
  f32x8 acc0 = z, acc1 = z, acc2 = z, acc3 = z;

  for (int kc = 0; kc < NN / 128; ++kc) {
    __syncthreads();
    for (int e = tid; e < 128 * HH; e += 64) {
      const int k = e >> 6, col = e & 63;
      pb[(k >> 1) * 128 + col * 2 + (k & 1)] = seq[(size_t)(kc * 128 + k) * HH + col];
    }
    __syncthreads();
    for (int kb = 0; kb < 128; kb += 4) {
      const f32x2 a = *(const f32x2*)(wm + (size_t)(rb + m) * NN + kc * 128 + kb + hi * 2);
      const float* pbase = pb + ((kb >> 1) + hi) * 128 + m * 2;
      acc0 = wmma4(a, *(const f32x2*)(pbase + 0 * 32), acc0);
      acc1 = wmma4(a, *(const f32x2*)(pbase + 1 * 32), acc1);
      acc2 = wmma4(a, *(const f32x2*)(pbase + 2 * 32), acc2);
      acc3 = wmma4(a, *(const f32x2*)(pbase + 3 * 32), acc3);
    }
  }

#pragma unroll
  for (int r = 0; r < 8; ++r) {
    const size_t rowoff = (size_t)(rb + r + 8 * hi) * HH + m;
    out[rowoff + 0 * 16] = acc0[r];
    out[rowoff + 1 * 16] = acc1[r];
    out[rowoff + 2 * 16] = acc2[r];
    out[rowoff + 3 * 16] = acc3[r];
  }
}

// ---------------------------------------------------------------------------
// Kernel 6: prediction[i] = leaky([seq | prop] @ w4 + b4)
// ---------------------------------------------------------------------------
__global__ void final_kernel(const float* __restrict__ seq,
                             const float* __restrict__ prop,
                             const float* __restrict__ w4, const float* __restrict__ b4,
                             float* __restrict__ out)
{
  const int i = blockIdx.x * 256 + threadIdx.x;
  if (i >= NN) return;
  float acc = b4[0];
#pragma unroll 8
  for (int k = 0; k < HH; ++k) acc += seq[(size_t)i * HH + k] * w4[k];
#pragma unroll 8
  for (int k = 0; k < HH; ++k) acc += prop[(size_t)i * HH + k] * w4[HH + k];
  out[i] = leaky(acc);
}

// ---------------------------------------------------------------------------
extern "C" void kernel_launch(void* const* d_in, const int* in_sizes, int n_in,
                              void* d_out, int out_size, void* d_ws, size_t ws_size,
                              hipStream_t stream) {
  const float* x        = (const float*)d_in[0];   // [N,T,D]
  const float* relation = (const float*)d_in[1];   // [N,N,R]
  const float* rel_mask = (const float*)d_in[2];   // [N,N]
  const float* W_ih     = (const float*)d_in[3];   // [4H,D]
  const float* W_hh     = (const float*)d_in[4];   // [4H,H]
  const float* b_ih     = (const float*)d_in[5];   // [4H]
  const float* b_hh     = (const float*)d_in[6];   // [4H]
  const float* w1       = (const float*)d_in[7];   // [R,1]
  const float* b1       = (const float*)d_in[8];
  const float* w2       = (const float*)d_in[9];   // [H,1]
  const float* b2       = (const float*)d_in[10];
  const float* w3       = (const float*)d_in[11];  // [H,1]
  const float* b3       = (const float*)d_in[12];
  const float* w4       = (const float*)d_in[13];  // [2H,1]
  const float* b4       = (const float*)d_in[14];
  float* out = (float*)d_out;

  float* ws      = (float*)d_ws;
  float* seq_emb = ws;                       // N*H
  float* prop    = ws + (size_t)NN * HH;     // N*H
  float* ht      = prop + (size_t)NN * HH;   // N
  float* logits  = ht + NN;                  // N*N

  lstm_kernel<<<NN / 32, 64, 0, stream>>>(x, W_ih, W_hh, b_ih, b_hh, seq_emb);
  ht_kernel<<<NN / 256, 256, 0, stream>>>(seq_emb, w2, b2, w3, b3, ht);
  rel_kernel<<<(NN * NN) / 256, 256, 0, stream>>>(relation, rel_mask, w1, b1, ht, logits);
  softmax_kernel<<<NN / 256, 256, 0, stream>>>(logits);
  prop_kernel<<<NN / 32, 64, 0, stream>>>(logits, seq_emb, prop);
  final_kernel<<<NN / 256, 256, 0, stream>>>(seq_emb, prop, w4, b4, out);
}